// GeneralSequentialImportanceSampler_78864189489649
// MI455X (gfx1250) — compile-verified
//
#include <hip/hip_runtime.h>
#include <hip/hip_bf16.h>
#include <math.h>

// ---------------- problem constants ----------------
#define NP      1048576
#define DIM     32
#define SIGMA   0.5f
#define TAU     0.3f
// em_lp = -0.5*sumsq/TAU^2 - D*(log(TAU)+0.5*log(2pi))
#define EM_SCALE  (-5.555555555f)        // -0.5 / (0.3*0.3)
#define EM_CONST  (9.120966758804290f)   // -32*(ln0.3 + 0.5*ln(2pi)) = +9.1209667...

typedef float v2f __attribute__((ext_vector_type(2)));
typedef float v8f __attribute__((ext_vector_type(8)));

__device__ __forceinline__ v8f wmma_f32(v2f a, v2f b, v8f c) {
    // V_WMMA_F32_16X16X4_F32 : D = A(16x4) x B(4x16) + C(16x16)
    return __builtin_amdgcn_wmma_f32_16x16x4_f32(
        false, a, false, b, (short)0, c, false, false);
}

// ================= reduction: max(log_w) =================
__global__ __launch_bounds__(256) void k_redmax(const float* __restrict__ lw,
                                                float* __restrict__ part) {
    __shared__ float s[256];
    const int t = threadIdx.x;
    const int base = blockIdx.x * 1024;
    float m = -INFINITY;
#pragma unroll
    for (int j = 0; j < 4; ++j) m = fmaxf(m, lw[base + j * 256 + t]);
    s[t] = m; __syncthreads();
    for (int off = 128; off; off >>= 1) {
        if (t < off) s[t] = fmaxf(s[t], s[t + off]);
        __syncthreads();
    }
    if (t == 0) part[blockIdx.x] = s[0];
}

__global__ __launch_bounds__(256) void k_finmax(const float* __restrict__ part,
                                                float* __restrict__ scal) {
    __shared__ float s[256];
    const int t = threadIdx.x;
    float m = -INFINITY;
#pragma unroll
    for (int j = 0; j < 4; ++j) m = fmaxf(m, part[j * 256 + t]);
    s[t] = m; __syncthreads();
    for (int off = 128; off; off >>= 1) {
        if (t < off) s[t] = fmaxf(s[t], s[t + off]);
        __syncthreads();
    }
    if (t == 0) scal[0] = s[0];
}

// ====== reduction: sum exp(lw-m), sum exp(2(lw-m)) ======
__global__ __launch_bounds__(256) void k_redsum(const float* __restrict__ lw,
                                                const float* __restrict__ scal,
                                                float* __restrict__ p1,
                                                float* __restrict__ p2) {
    __shared__ float s1[256], s2[256];
    const int t = threadIdx.x;
    const int base = blockIdx.x * 1024;
    const float m = scal[0];
    float a = 0.f, b = 0.f;
#pragma unroll
    for (int j = 0; j < 4; ++j) {
        float x = lw[base + j * 256 + t] - m;
        a += __expf(x);
        b += __expf(2.0f * x);
    }
    s1[t] = a; s2[t] = b; __syncthreads();
    for (int off = 128; off; off >>= 1) {
        if (t < off) { s1[t] += s1[t + off]; s2[t] += s2[t + off]; }
        __syncthreads();
    }
    if (t == 0) { p1[blockIdx.x] = s1[0]; p2[blockIdx.x] = s2[0]; }
}

__global__ __launch_bounds__(256) void k_finsum(const float* __restrict__ p1,
                                                const float* __restrict__ p2,
                                                float* __restrict__ scal,
                                                float* __restrict__ out_ess) {
    __shared__ float s1[256], s2[256];
    const int t = threadIdx.x;
    float a = 0.f, b = 0.f;
#pragma unroll
    for (int j = 0; j < 4; ++j) { a += p1[j * 256 + t]; b += p2[j * 256 + t]; }
    s1[t] = a; s2[t] = b; __syncthreads();
    for (int off = 128; off; off >>= 1) {
        if (t < off) { s1[t] += s1[t + off]; s2[t] += s2[t + off]; }
        __syncthreads();
    }
    if (t == 0) {
        float S1 = s1[0], S2 = s2[0];
        float m = scal[0];
        scal[1] = m + logf(S1);                         // log_sum_w
        float ess = (S1 * S1) / (S2 * (float)NP);       // exp(2lsw - lsw2)/N
        scal[2] = ess;
        out_ess[0] = ess;
    }
}

// ============ prefix sum of softmax weights ============
__global__ __launch_bounds__(256) void k_scan1(const float* __restrict__ lw,
                                               const float* __restrict__ scal,
                                               float* __restrict__ cum,
                                               float* __restrict__ bsums) {
    __shared__ float wt[8], wo[8];
    const int t = threadIdx.x;
    const int lane = t & 31, wave = t >> 5;
    const int gid = blockIdx.x * 256 + t;
    const float lsw = scal[1];
    const float w = __expf(lw[gid] - lsw);
    float v = w;
#pragma unroll
    for (int d = 1; d < 32; d <<= 1) {
        float n = __shfl_up(v, d, 32);
        if (lane >= d) v += n;
    }
    if (lane == 31) wt[wave] = v;
    __syncthreads();
    if (t == 0) {
        float r = 0.f;
#pragma unroll
        for (int i = 0; i < 8; ++i) { wo[i] = r; r += wt[i]; }
    }
    __syncthreads();
    const float incl = v + wo[wave];
    cum[gid] = incl;
    if (t == 255) bsums[blockIdx.x] = incl;
}

__global__ __launch_bounds__(256) void k_scan2(float* __restrict__ bsums) {
    __shared__ float wt[8], wexcl[8], chunk_total;
    const int t = threadIdx.x;
    const int lane = t & 31, wave = t >> 5;
    float carry = 0.f;
    for (int c = 0; c < 16; ++c) {
        const float s = bsums[c * 256 + t];
        float v = s;
#pragma unroll
        for (int d = 1; d < 32; d <<= 1) {
            float n = __shfl_up(v, d, 32);
            if (lane >= d) v += n;
        }
        if (lane == 31) wt[wave] = v;
        __syncthreads();
        if (t == 0) {
            float r = 0.f;
#pragma unroll
            for (int i = 0; i < 8; ++i) { float x = wt[i]; wexcl[i] = r; r += x; }
            chunk_total = r;
        }
        __syncthreads();
        const float excl = carry + wexcl[wave] + v - s;   // exclusive block offset
        bsums[c * 256 + t] = excl;
        carry += chunk_total;
        __syncthreads();
    }
}

__global__ __launch_bounds__(256) void k_scan3(float* __restrict__ cum,
                                               const float* __restrict__ bsums) {
    const int gid = blockIdx.x * 256 + threadIdx.x;
    cum[gid] += bsums[blockIdx.x];
}

// ==== systematic resampling: searchsorted + ESS gate ====
__global__ __launch_bounds__(256) void k_search(const float* __restrict__ cum,
                                                const float* __restrict__ scal,
                                                const float* __restrict__ u,
                                                int* __restrict__ idx) {
    const int i = blockIdx.x * 256 + threadIdx.x;
    const float pos = ((float)i + u[0]) * (1.0f / (float)NP);
    int lo = 0, hi = NP;
    while (lo < hi) {
        int mid = (lo + hi) >> 1;
        if (cum[mid] < pos) lo = mid + 1; else hi = mid;
    }
    int r = lo < (NP - 1) ? lo : (NP - 1);
    idx[i] = (scal[2] < 0.5f) ? r : i;
}

// ======== fused: gather + X@A^T + noise + next@C^T + em_lp ========
// One wave per 16-row tile; 8 waves (128 rows) per block.
__global__ __launch_bounds__(256) void k_main(const float* __restrict__ log_w,
                                              const float* __restrict__ particles,
                                              const float* __restrict__ obs,
                                              const float* __restrict__ eps,
                                              const float* __restrict__ A,
                                              const float* __restrict__ C,
                                              const int* __restrict__ idx,
                                              float* __restrict__ out_logw,
                                              float* __restrict__ out_next) {
    __shared__ float lds[8][16 * 34];   // padded stride 34: even (b64-aligned) + conflict-free
    const int lane = threadIdx.x & 31;
    const int wave = threadIdx.x >> 5;
    const int tile = blockIdx.x * 8 + wave;
    const int base = tile * 16;
    const int nl = lane & 15;          // row (A-frag) / column (B,C,D-frag)
    const int half = lane >> 4;        // 0 or 1
    const int kb = half * 2;           // K sub-pair per half (ISA 16x4 layout)

    // --- B-fragments for mu = X @ A^T : B[k][n] = A[n][k] (contiguous f2 loads) ---
    v2f bA0[8], bA1[8];
#pragma unroll
    for (int kk = 0; kk < 8; ++kk) {
        bA0[kk] = *(const v2f*)(A + nl * 32 + 4 * kk + kb);
        bA1[kk] = *(const v2f*)(A + (nl + 16) * 32 + 4 * kk + kb);
    }

    // --- gather resampled X rows directly into 16x4 A-frag layout ---
    const int row = idx[base + nl];
    const float* px = particles + (size_t)row * 32 + kb;
    v2f aX[8];
#pragma unroll
    for (int kk = 0; kk < 8; ++kk) aX[kk] = *(const v2f*)(px + 4 * kk);

    v8f acc0 = {}, acc1 = {};
#pragma unroll
    for (int kk = 0; kk < 8; ++kk) {
        acc0 = wmma_f32(aX[kk], bA0[kk], acc0);   // cols 0..15
        acc1 = wmma_f32(aX[kk], bA1[kk], acc1);   // cols 16..31
    }

    // --- next = mu + SIGMA*eps ; store + stage transpose through LDS ---
    // (trans_lp == prop_lp for bootstrap proposal -> they cancel exactly)
    float* ldsw = &lds[wave][0];
#pragma unroll
    for (int v = 0; v < 8; ++v) {
        const int m = v + 8 * half;                 // D-frag row
        const size_t g = (size_t)(base + m) * 32;
        float n0 = fmaf(SIGMA, eps[g + nl], acc0[v]);
        float n1 = fmaf(SIGMA, eps[g + nl + 16], acc1[v]);
        out_next[g + nl] = n0;
        out_next[g + nl + 16] = n1;
        ldsw[m * 34 + nl] = n0;
        ldsw[m * 34 + nl + 16] = n1;
    }
    __syncthreads();

    // --- em_mu = next @ C^T ---
    v2f aN[8], bC0[8], bC1[8];
#pragma unroll
    for (int kk = 0; kk < 8; ++kk) {
        aN[kk]  = *(const v2f*)(&ldsw[nl * 34 + 4 * kk + kb]);
        bC0[kk] = *(const v2f*)(C + nl * 32 + 4 * kk + kb);
        bC1[kk] = *(const v2f*)(C + (nl + 16) * 32 + 4 * kk + kb);
    }
    v8f acc2 = {}, acc3 = {};
#pragma unroll
    for (int kk = 0; kk < 8; ++kk) {
        acc2 = wmma_f32(aN[kk], bC0[kk], acc2);
        acc3 = wmma_f32(aN[kk], bC1[kk], acc3);
    }

    // --- em_lp: row-wise sumsq of (obs - em_mu), reduced across 16-lane halves ---
    const float o0 = obs[nl], o1 = obs[nl + 16];
    float t[8];
#pragma unroll
    for (int v = 0; v < 8; ++v) {
        float r0 = o0 - acc2[v];
        float r1 = o1 - acc3[v];
        float s = fmaf(r0, r0, r1 * r1);
        s += __shfl_xor(s, 1, 32);
        s += __shfl_xor(s, 2, 32);
        s += __shfl_xor(s, 4, 32);
        s += __shfl_xor(s, 8, 32);   // sums lanes within each 16-lane half
        t[v] = s;
    }
    if (nl == 0) {                    // lane 0 -> rows 0..7, lane 16 -> rows 8..15
#pragma unroll
        for (int v = 0; v < 8; ++v) {
            const int m = v + 8 * half;
            const float em = fmaf(EM_SCALE, t[v], EM_CONST);
            out_logw[base + m] = log_w[base + m] + em;
        }
    }
}

// ======================= host launch =======================
extern "C" void kernel_launch(void* const* d_in, const int* in_sizes, int n_in,
                              void* d_out, int out_size, void* d_ws, size_t ws_size,
                              hipStream_t stream) {
    const float* log_w     = (const float*)d_in[0];
    const float* particles = (const float*)d_in[1];
    const float* obs       = (const float*)d_in[2];
    const float* eps       = (const float*)d_in[3];
    const float* u         = (const float*)d_in[4];
    const float* A         = (const float*)d_in[5];
    const float* C         = (const float*)d_in[6];

    float* out      = (float*)d_out;
    float* out_logw = out;                       // [N]
    float* out_next = out + NP;                  // [N*32]
    float* out_ess  = out + NP + (size_t)NP * 32; // [1]

    float* ws    = (float*)d_ws;
    float* cum   = ws;                  // N floats
    int*   idx   = (int*)(ws + NP);     // N ints
    float* p1    = ws + 2 * (size_t)NP; // 1024
    float* p2    = p1 + 1024;           // 1024
    float* bsums = p2 + 1024;           // 4096
    float* scal  = bsums + 4096;        // [0]=max, [1]=log_sum_w, [2]=ess

    // ESS
    k_redmax<<<1024, 256, 0, stream>>>(log_w, p1);
    k_finmax<<<1, 256, 0, stream>>>(p1, scal);
    k_redsum<<<1024, 256, 0, stream>>>(log_w, scal, p1, p2);
    k_finsum<<<1, 256, 0, stream>>>(p1, p2, scal, out_ess);

    // softmax cumsum (3-stage scan) + systematic resample indices
    k_scan1<<<4096, 256, 0, stream>>>(log_w, scal, cum, bsums);
    k_scan2<<<1, 256, 0, stream>>>(bsums);
    k_scan3<<<4096, 256, 0, stream>>>(cum, bsums);
    k_search<<<4096, 256, 0, stream>>>(cum, scal, u, idx);

    // fused gather + GEMM(A) + noise + GEMM(C) + weight update
    k_main<<<NP / 128, 256, 0, stream>>>(log_w, particles, obs, eps, A, C,
                                         idx, out_logw, out_next);
}